// PlenoxelModel_1211180777946
// MI455X (gfx1250) — compile-verified
//
#include <hip/hip_runtime.h>
#include <math.h>

#define GXD 128
#define GYD 128
#define GZD 128
#define VD 28
#define NUM_RAYS 8192
#define NUM_SAMPLES 128

typedef float v2f __attribute__((ext_vector_type(2)));
typedef float v8f __attribute__((ext_vector_type(8)));

// P[k][col] of the 28x16 projection matrix:
//   col 0 : sigma extractor (k==0)
//   col 1 : basis[k-1]  for k in [1,9]
//   col 2 : basis[k-10] for k in [10,18]
//   col 3 : basis[k-19] for k in [19,27]
//   col>3 : 0
// k is compile-time at all call sites (unrolled loops).
__device__ __forceinline__ float Pv(int k, int col, const float* bb) {
    float c1 = (k >= 1  && k <= 9 ) ? bb[k - 1]  : 0.0f;
    float c2 = (k >= 10 && k <= 18) ? bb[k - 10] : 0.0f;
    float c3 = (k >= 19 && k <= 27) ? bb[k - 19] : 0.0f;
    float r = 0.0f;
    r = (col == 0) ? ((k == 0) ? 1.0f : 0.0f) : r;
    r = (col == 1) ? c1 : r;
    r = (col == 2) ? c2 : r;
    r = (col == 3) ? c3 : r;
    return r;
}

// One wave: builds the per-lane packed WMMA-B operands for all 7 K-chunks.
// Chunk i, lane l holds {P[klo][col], P[klo+1][col]} with
//   klo = (l<16) ? 4i : 4i+2, col = l&15   (ISA 16x4 f32 A/B lane striping).
__global__ __launch_bounds__(32)
void build_proj_kernel(const float* __restrict__ view, float2* __restrict__ bw) {
    const int lane = threadIdx.x;
    const bool lo  = lane < 16;
    const int col  = lane & 15;

    const float theta = view[0], phi = view[1];
    const float st = sinf(theta), ct = cosf(theta);
    const float sp = sinf(phi),   cp = cosf(phi);
    float bb[9];
    bb[0] = 0.28209479177387814f;
    bb[1] = 0.4886025119029199f * st * sp;
    bb[2] = 0.4886025119029199f * ct;
    bb[3] = 0.4886025119029199f * st * cp;
    bb[4] = 1.0925484305920792f * st * cp * st * sp;
    bb[5] = 1.0925484305920792f * st * sp * ct;
    bb[6] = 0.31539156525252005f * (3.0f * ct * ct - 1.0f);
    bb[7] = 1.0925484305920792f * st * cp * ct;
    bb[8] = 0.5462742152960396f * ((st * cp) * (st * cp) - (st * sp) * (st * sp));

#pragma unroll
    for (int i = 0; i < 7; ++i) {
        const float x_lo = Pv(4 * i + 0, col, bb);
        const float x_hi = Pv(4 * i + 2, col, bb);
        const float y_lo = Pv(4 * i + 1, col, bb);
        const float y_hi = Pv(4 * i + 3, col, bb);
        float2 v;
        v.x = lo ? x_lo : x_hi;
        v.y = lo ? y_lo : y_hi;
        bw[i * 32 + lane] = v;
    }
}

__global__ __launch_bounds__(NUM_SAMPLES)
void plenoxel_render_kernel(const float* __restrict__ grid,
                            const float* __restrict__ positions,
                            const float* __restrict__ distances,
                            const float2* __restrict__ bw,
                            float* __restrict__ out) {
    const int ray  = blockIdx.x;
    const int s    = threadIdx.x;          // sample id within ray, 0..127
    const int lane = s & 31;
    const int wave = s >> 5;
    const int wavebase = s & ~31;          // first sample of this wave
    const bool lo  = lane < 16;
    const int col  = lane & 15;

    // ---------------- trilinear gather + blend of 28-dim features ----------
    const int pidx = (ray * NUM_SAMPLES + s) * 3;
    const float px = positions[pidx + 0];
    const float py = positions[pidx + 1];
    const float pz = positions[pidx + 2];
    const float dd = distances[ray * NUM_SAMPLES + s];

    const float fx = floorf(px), fy = floorf(py), fz = floorf(pz);
    const int x0 = (int)fx, y0 = (int)fy, z0 = (int)fz;
    const float xd = px - fx, yd = py - fy, zd = pz - fz;

    float f[VD];

    // corner 0 seeds f[] (no zero-init + dead adds)
    {
        const float w = (1.0f - xd) * (1.0f - yd) * (1.0f - zd);
        const float4* g4 = reinterpret_cast<const float4*>(
            grid + (size_t)(((x0 * GYD) + y0) * GZD + z0) * VD);
#pragma unroll
        for (int q = 0; q < 7; ++q) {
            float4 v = g4[q];
            f[4 * q + 0] = w * v.x;
            f[4 * q + 1] = w * v.y;
            f[4 * q + 2] = w * v.z;
            f[4 * q + 3] = w * v.w;
        }
    }

#pragma unroll
    for (int c = 1; c < 8; ++c) {
        const int cx = x0 + (c & 1);
        const int cy = y0 + ((c >> 1) & 1);
        const int cz = z0 + ((c >> 2) & 1);
        const float w = ((c & 1)        ? xd : 1.0f - xd) *
                        (((c >> 1) & 1) ? yd : 1.0f - yd) *
                        (((c >> 2) & 1) ? zd : 1.0f - zd);
        const float4* g4 = reinterpret_cast<const float4*>(
            grid + (size_t)(((cx * GYD) + cy) * GZD + cz) * VD);
#pragma unroll
        for (int q = 0; q < 7; ++q) {
            float4 v = g4[q];
            f[4 * q + 0] = fmaf(w, v.x, f[4 * q + 0]);
            f[4 * q + 1] = fmaf(w, v.y, f[4 * q + 1]);
            f[4 * q + 2] = fmaf(w, v.z, f[4 * q + 2]);
            f[4 * q + 3] = fmaf(w, v.w, f[4 * q + 3]);
        }
    }

    // ---------------- WMMA projection: D(16x16) = feats(16x28) x P(28x16) --
    // Two tiles per wave: rows = samples wavebase+0..15 and wavebase+16..31.
    // A layout (ISA 7.12.2, 32-bit 16x4): VGPR0 lanes0-15=K0 / lanes16-31=K2,
    // VGPR1 = K1 / K3.  B operands precomputed per-lane in bw[].
    v8f acc0 = {0.f, 0.f, 0.f, 0.f, 0.f, 0.f, 0.f, 0.f};
    v8f acc1 = {0.f, 0.f, 0.f, 0.f, 0.f, 0.f, 0.f, 0.f};

#pragma unroll
    for (int i = 0; i < 7; ++i) {
        // partner-lane (lane ^ 16) feature values
        const float sA = __shfl_xor(f[4 * i + 0], 16, 32);
        const float sB = __shfl_xor(f[4 * i + 1], 16, 32);
        const float sC = __shfl_xor(f[4 * i + 2], 16, 32);
        const float sD = __shfl_xor(f[4 * i + 3], 16, 32);

        v2f a0, a1;
        // tile0: rows = samples wavebase+0..15
        a0.x = lo ? f[4 * i + 0] : sC;   // lanes>=16 carry K2/K3 of sample lane-16
        a0.y = lo ? f[4 * i + 1] : sD;
        // tile1: rows = samples wavebase+16..31
        a1.x = lo ? sA : f[4 * i + 2];   // lanes<16 carry K0/K1 of sample lane+16
        a1.y = lo ? sB : f[4 * i + 3];

        const float2 bl = bw[i * 32 + lane];
        v2f bv; bv.x = bl.x; bv.y = bl.y;

        acc0 = __builtin_amdgcn_wmma_f32_16x16x4_f32(
            false, a0, false, bv, (short)0, acc0, false, false);
        acc1 = __builtin_amdgcn_wmma_f32_16x16x4_f32(
            false, a1, false, bv, (short)0, acc1, false, false);
    }

    // ---------------- scatter (sigma, r, g, b) back to sample owners -------
    __shared__ float ldsOut[NUM_SAMPLES][4];   // 2 KB
    // C/D layout: VGPR r -> lanes0-15: (M=r, N=lane); lanes16-31: (M=r+8, N=lane-16)
#pragma unroll
    for (int r = 0; r < 8; ++r) {
        const int M = lo ? r : (r + 8);
        if (col < 4) {
            ldsOut[wavebase + M][col]      = acc0[r];
            ldsOut[wavebase + 16 + M][col] = acc1[r];
        }
    }
    __syncthreads();

    const float sigma = ldsOut[s][0];
    const float rc    = ldsOut[s][1];
    const float gc    = ldsOut[s][2];
    const float bc    = ldsOut[s][3];

    // ---------------- attenuation + inclusive cumsum (wave scan) -----------
    const float att = expf(-sigma * dd);

    // inclusive scan within wave (5 shuffle steps)
    float v = att;
#pragma unroll
    for (int d = 1; d < 32; d <<= 1) {
        const float t = __shfl_up(v, d, 32);
        if (lane >= d) v += t;
    }
    // cross-wave offsets
    __shared__ float wsum[4];
    if (lane == 31) wsum[wave] = v;
    __syncthreads();
    float base = 0.0f;
    for (int w = 0; w < wave; ++w) base += wsum[w];
    const float run = v + base;                 // inclusive cumsum of att
    const float wgt = run * (1.0f - att);       // transmittance * alpha

    // ---------------- per-ray reduction of weights * rgb -------------------
    float cr = wgt * rc, cg = wgt * gc, cb = wgt * bc;
#pragma unroll
    for (int d = 16; d > 0; d >>= 1) {
        cr += __shfl_xor(cr, d, 32);
        cg += __shfl_xor(cg, d, 32);
        cb += __shfl_xor(cb, d, 32);
    }
    __shared__ float part[4][3];
    if (lane == 0) {
        part[wave][0] = cr;
        part[wave][1] = cg;
        part[wave][2] = cb;
    }
    __syncthreads();
    if (s < 3) {
        out[ray * 3 + s] = part[0][s] + part[1][s] + part[2][s] + part[3][s];
    }
}

extern "C" void kernel_launch(void* const* d_in, const int* in_sizes, int n_in,
                              void* d_out, int out_size, void* d_ws, size_t ws_size,
                              hipStream_t stream) {
    (void)in_sizes; (void)n_in; (void)ws_size; (void)out_size;
    const float* grid      = (const float*)d_in[0];   // [128,128,128,28]
    const float* positions = (const float*)d_in[1];   // [8192,128,3]
    const float* distances = (const float*)d_in[2];   // [8192,128]
    const float* view      = (const float*)d_in[3];   // [2]
    float* out             = (float*)d_out;           // [8192,3]
    float2* bw             = (float2*)d_ws;           // 7*32 packed B operands

    build_proj_kernel<<<1, 32, 0, stream>>>(view, bw);
    plenoxel_render_kernel<<<NUM_RAYS, NUM_SAMPLES, 0, stream>>>(
        grid, positions, distances, bw, out);
}